// AttnBlock_31035433680983
// MI455X (gfx1250) — compile-verified
//
#include <hip/hip_runtime.h>
#include <math.h>
#include <stdint.h>

// ---------------------------------------------------------------------------
// Equivariant attention block for MI455X (gfx1250, wave32).
// Radial MLPs (3->16->16->{4,12}) run on V_WMMA_F32_16X16X4_F32 in a
// transposed formulation: h^T = W^T x X^T, 16 edges per WMMA tile.
//   A (16x4 f32): lanes 0-15 = M rows, vgpr0/1 = K0/K1; lanes 16-31 = K2/K3.
//   B (4x16 f32): lanes 0-15 = cols N, vgpr0/1 = K0/K1; lanes 16-31 = K2/K3.
//   D (16x16 f32): lane holds col N = lane&15, rows M = vgpr + 8*(lane>=16).
// => per edge, hidden features split across a lane pair {L, L+16}: LayerNorm
//    needs one shfl_xor(16); layer-2/3 K-chunks need <=2 shfl_xor(16) each.
// Per-edge wj tensors (136 B/edge, the bulk of HBM traffic) are staged into
// LDS with GLOBAL_LOAD_ASYNC_TO_LDS_B128 (ASYNCcnt) issued *before* the
// WMMA chain, so the 72 WMMAs hide the gather latency; s_wait_asynccnt 0
// before the epilogue consumes them from LDS.
// Softmax over destination nodes via float atomic max/add.
// ---------------------------------------------------------------------------

typedef __attribute__((ext_vector_type(2))) float v2f;
typedef __attribute__((ext_vector_type(8))) float v8f;

__device__ __forceinline__ v8f wmma4(v2f a, v2f b, v8f c) {
  // 8 args: (neg_a, A, neg_b, B, c_mod, C, reuse_a, reuse_b)
  return __builtin_amdgcn_wmma_f32_16x16x4_f32(false, a, false, b, (short)0, c,
                                               false, false);
}

__device__ __forceinline__ float xshf(float x) { return __shfl_xor(x, 16, 32); }

// float atomic max via signed-int max / unsigned-int min trick (init = -inf).
__device__ __forceinline__ void atomicMaxFloat(float* addr, float val) {
  if (val >= 0.0f)
    atomicMax((int*)addr, __float_as_int(val));
  else
    atomicMin((unsigned int*)addr, __float_as_uint(val));
}

// LayerNorm (+bias first, ReLU after) over 16 features split across lane pair.
__device__ __forceinline__ void ln_relu(float* h, const float* rb,
                                        const float* g, const float* be,
                                        int hi) {
  float s = 0.f, ss = 0.f;
#pragma unroll
  for (int r = 0; r < 8; ++r) {
    h[r] += rb[r + 8 * hi];
    s += h[r];
    ss += h[r] * h[r];
  }
  s += xshf(s);
  ss += xshf(ss);
  const float mu = s * 0.0625f;
  const float var = ss * 0.0625f - mu * mu;
  const float inv = rsqrtf(var + 1e-5f);
#pragma unroll
  for (int r = 0; r < 8; ++r) {
    const int n = r + 8 * hi;
    const float x = (h[r] - mu) * inv * g[n] + be[n];
    h[r] = fmaxf(x, 0.f);
  }
}

// Build B operand (rows K = 4*KC+2*hi+{0,1} of h^T) from D-layout registers.
template <int KC>
__device__ __forceinline__ v2f bchunk(const float* h, int hi) {
  v2f b;
  if (KC < 2) {
    const float t0 = xshf(h[4 * KC + 2]);
    const float t1 = xshf(h[4 * KC + 3]);
    b.x = hi ? t0 : h[4 * KC + 0];
    b.y = hi ? t1 : h[4 * KC + 1];
  } else {
    const float t0 = xshf(h[4 * KC - 8]);
    const float t1 = xshf(h[4 * KC - 7]);
    b.x = hi ? h[4 * KC - 6] : t0;
    b.y = hi ? h[4 * KC - 5] : t1;
  }
  return b;
}

// One 16-edge MLP pass; lanes with (hi==keep) take ownership of P outputs.
template <int P>
__device__ __forceinline__ void run_mlp(v2f a1, const v2f* a2, const v2f* a3,
                                        v2f bin, const float* rb1,
                                        const float* g1, const float* be1,
                                        const float* rb2, const float* g2,
                                        const float* be2, const float* b3,
                                        int hi, bool keep, float* R) {
  const v8f zero = {0.f, 0.f, 0.f, 0.f, 0.f, 0.f, 0.f, 0.f};
  // layer 1: K=3 (padded to 4)
  v8f c = wmma4(a1, bin, zero);
  float h[8];
#pragma unroll
  for (int r = 0; r < 8; ++r) h[r] = c[r];
  ln_relu(h, rb1, g1, be1, hi);
  // layer 2: K=16 as 4 accumulating K=4 WMMAs
  v8f c2 = zero;
  c2 = wmma4(a2[0], bchunk<0>(h, hi), c2);
  c2 = wmma4(a2[1], bchunk<1>(h, hi), c2);
  c2 = wmma4(a2[2], bchunk<2>(h, hi), c2);
  c2 = wmma4(a2[3], bchunk<3>(h, hi), c2);
  float h2[8];
#pragma unroll
  for (int r = 0; r < 8; ++r) h2[r] = c2[r];
  ln_relu(h2, rb2, g2, be2, hi);
  // layer 3: 16 -> P (rows M >= P are zero in A)
  v8f c3 = zero;
  c3 = wmma4(a3[0], bchunk<0>(h2, hi), c3);
  c3 = wmma4(a3[1], bchunk<1>(h2, hi), c3);
  c3 = wmma4(a3[2], bchunk<2>(h2, hi), c3);
  c3 = wmma4(a3[3], bchunk<3>(h2, hi), c3);
  float d[8];
#pragma unroll
  for (int r = 0; r < 8; ++r) d[r] = c3[r];
  // bias (branchless)
#pragma unroll
  for (int r = 0; r < 8; ++r) {
    const int n = r + 8 * hi;
    const int idx = n < P ? n : P - 1;
    d[r] += (n < P ? 1.f : 0.f) * b3[idx];
  }
  // redistribute so owner lane holds all P outputs of its edge
  constexpr int P1 = (P < 8) ? P : 8;
#pragma unroll
  for (int n = 0; n < P1; ++n) {
    const float t = xshf(d[n]);
    const float val = hi ? t : d[n];
    R[n] = keep ? val : R[n];
  }
#pragma unroll
  for (int n = 8; n < P; ++n) {
    const float t = xshf(d[n - 8]);
    const float val = hi ? d[n - 8] : t;
    R[n] = keep ? val : R[n];
  }
}

template <int P>
__device__ __forceinline__ void mlp_full(
    int i, int M, int hi, const v2f* bin, const float* rW1, const float* rb1,
    const float* g1, const float* be1, const float* rW2, const float* rb2,
    const float* g2, const float* be2, const float* W3, const float* b3,
    float* R) {
  const float* W1 = rW1 + i * 48;  // [3,16]
  v2f a1;
  a1.x = W1[(2 * hi) * 16 + M];
  a1.y = hi ? 0.f : W1[16 + M];  // K=3 row is zero pad
  const float* W2 = rW2 + i * 256;  // [16,16]
  v2f a2[4], a3[4];
#pragma unroll
  for (int kc = 0; kc < 4; ++kc) {
    const int K = 4 * kc + 2 * hi;
    a2[kc].x = W2[K * 16 + M];
    a2[kc].y = W2[(K + 1) * 16 + M];
    const float m = (M < P) ? 1.f : 0.f;
    const int Mi = (M < P) ? M : 0;
    a3[kc].x = m * W3[K * P + Mi];
    a3[kc].y = m * W3[(K + 1) * P + Mi];
  }
  const float* Rb1 = rb1 + i * 16;
  const float* G1 = g1 + i * 16;
  const float* Be1 = be1 + i * 16;
  const float* Rb2 = rb2 + i * 16;
  const float* G2 = g2 + i * 16;
  const float* Be2 = be2 + i * 16;
#pragma unroll
  for (int t = 0; t < 2; ++t)
    run_mlp<P>(a1, a2, a3, bin[t], Rb1, G1, Be1, Rb2, G2, Be2, b3, hi,
               (hi == t), R);
}

__global__ void init_kernel(float* mbuf, float* sbuf, int N) {
  const int n = blockIdx.x * blockDim.x + threadIdx.x;
  if (n < N) {
    mbuf[n] = -INFINITY;
    sbuf[n] = 0.f;
  }
}

// Per-wave LDS staging layout (dword offsets inside a 1088-dword wave slab):
//   wj00: [0,32)   wj01: [32,128)   wj10: [128,224)   wj11: [224,1088)
#define SLAB_DW 1088
#define SLAB_BYTES 4352u

__global__ __launch_bounds__(256) void edge_dot_kernel(
    const float* __restrict__ f0, const float* __restrict__ f1,
    const float* __restrict__ dist, const int* __restrict__ ui,
    const int* __restrict__ vi, const float* __restrict__ wq,
    const float* __restrict__ wj00, const float* __restrict__ wj01,
    const float* __restrict__ wj10, const float* __restrict__ wj11,
    const float* __restrict__ rW1, const float* __restrict__ rb1,
    const float* __restrict__ g1, const float* __restrict__ be1,
    const float* __restrict__ rW2, const float* __restrict__ rb2,
    const float* __restrict__ g2, const float* __restrict__ be2,
    const float* __restrict__ W300, const float* __restrict__ b300,
    const float* __restrict__ W301, const float* __restrict__ b301,
    const float* __restrict__ W310, const float* __restrict__ b310,
    const float* __restrict__ W311, const float* __restrict__ b311,
    float* __restrict__ dotbuf, float* __restrict__ mbuf, int E) {
  __shared__ float smem[8 * SLAB_DW];

  const int lane = threadIdx.x & 31;
  const int hi = lane >> 4;
  const int M = lane & 15;
  const int wv = threadIdx.x >> 5;
  const int tileBase = (blockIdx.x * 8 + wv) * 32;
  // clamped staging base so async addresses stay in-bounds on tail tiles
  const int tb = (tileBase + 32 <= E) ? tileBase : ((E > 32) ? (E - 32) : 0);

  // ---- async-stage wj tensors for this 32-edge tile into LDS -------------
  // 10 x global_load_async_to_lds_b128 per wave; all tile bases are
  // 16B-aligned (tb is a multiple of 32; 32*{4,12,108} bytes are 16-aligned).
  {
    const uint32_t lbase =
        (uint32_t)(uintptr_t)(&smem[0]) + (uint32_t)wv * SLAB_BYTES;
    const uint32_t c16 = (uint32_t)lane * 16u;
    if (lane < 8) {  // wj00: 32 dw = 8 x b128
      asm volatile("global_load_async_to_lds_b128 %0, %1, %2"
                   :
                   : "v"(lbase + c16), "v"((uint32_t)tb * 4u + c16), "s"(wj00)
                   : "memory");
    }
    if (lane < 24) {  // wj01 / wj10: 96 dw = 24 x b128 each
      asm volatile("global_load_async_to_lds_b128 %0, %1, %2"
                   :
                   : "v"(lbase + 128u + c16), "v"((uint32_t)tb * 12u + c16),
                     "s"(wj01)
                   : "memory");
      asm volatile("global_load_async_to_lds_b128 %0, %1, %2"
                   :
                   : "v"(lbase + 512u + c16), "v"((uint32_t)tb * 12u + c16),
                     "s"(wj10)
                   : "memory");
    }
#pragma unroll
    for (int r = 0; r < 6; ++r) {  // wj11: 864 dw = 216 x b128 (6 full rounds)
      const uint32_t o = (uint32_t)(r * 32 + lane) * 16u;
      asm volatile("global_load_async_to_lds_b128 %0, %1, %2"
                   :
                   : "v"(lbase + 896u + o), "v"((uint32_t)tb * 108u + o),
                     "s"(wj11)
                   : "memory");
    }
    if (lane < 24) {  // wj11 tail: 24 x b128
      const uint32_t o = (uint32_t)(192 + lane) * 16u;
      asm volatile("global_load_async_to_lds_b128 %0, %1, %2"
                   :
                   : "v"(lbase + 896u + o), "v"((uint32_t)tb * 108u + o),
                     "s"(wj11)
                   : "memory");
    }
  }

  // edge owned by this lane for the epilogue
  const int eo = tileBase + lane;
  const int eoc = (eo < E) ? eo : (E - 1);
  const int vo = vi[eoc];
  const float f0v0 = f0[vo * 2 + 0];
  const float f0v1 = f0[vo * 2 + 1];
  float f1v[6];
#pragma unroll
  for (int t = 0; t < 6; ++t) f1v[t] = f1[vo * 6 + t];

  // layer-1 B operands (radial input vec = [p0, p1, dist, 0]) per subtile
  v2f bin[2];
#pragma unroll
  for (int t = 0; t < 2; ++t) {
    int et = tileBase + t * 16 + M;
    et = (et < E) ? et : (E - 1);
    const int ue = ui[et], ve = vi[et];
    const float p0 = f0[ue * 2 + 0] * f0[ve * 2 + 0];
    const float p1 = f0[ue * 2 + 1] * f0[ve * 2 + 1];
    const float dd = dist[et];
    bin[t].x = hi ? dd : p0;
    bin[t].y = hi ? 0.f : p1;
  }

  float R00[4], R01[4], R10[4], R11[12];
#pragma unroll
  for (int j = 0; j < 4; ++j) { R00[j] = 0.f; R01[j] = 0.f; R10[j] = 0.f; }
#pragma unroll
  for (int j = 0; j < 12; ++j) R11[j] = 0.f;

  mlp_full<4>(0, M, hi, bin, rW1, rb1, g1, be1, rW2, rb2, g2, be2, W300, b300, R00);
  mlp_full<4>(1, M, hi, bin, rW1, rb1, g1, be1, rW2, rb2, g2, be2, W301, b301, R01);
  mlp_full<4>(2, M, hi, bin, rW1, rb1, g1, be1, rW2, rb2, g2, be2, W310, b310, R10);
  mlp_full<12>(3, M, hi, bin, rW1, rb1, g1, be1, rW2, rb2, g2, be2, W311, b311, R11);

  // wait for the async wj staging (latency hidden behind 72 WMMAs above)
  asm volatile("s_wait_asynccnt 0" ::: "memory");
  const float* wjL = &smem[wv * SLAB_DW];
  const int li = eoc - tb;  // in [0,32)

  // ---------------- per-edge epilogue (lane owns edge eo) ----------------
  // k_feat[o][0]  (degree-0 output)
  float kf00, kf01;
  {
    const float wj = wjL[li];
    kf00 = wj * (R00[0] * f0v0 + R00[1] * f0v1);
    kf01 = wj * (R00[2] * f0v0 + R00[3] * f0v1);
    float s0 = 0.f, s1 = 0.f;  // sum_k wj01[k]*f1v[i][k]
#pragma unroll
    for (int k = 0; k < 3; ++k) {
      const float w = wjL[32 + li * 3 + k];
      s0 += w * f1v[k];
      s1 += w * f1v[3 + k];
    }
    kf00 += R01[0] * s0 + R01[1] * s1;
    kf01 += R01[2] * s0 + R01[3] * s1;
  }
  // k_feat[o][1+l] (degree-1 outputs)
  float kf1[2][3];
  {
    const float s0 = R10[0] * f0v0 + R10[1] * f0v1;
    const float s1 = R10[2] * f0v0 + R10[3] * f0v1;
#pragma unroll
    for (int l = 0; l < 3; ++l) {
      const float w = wjL[128 + li * 3 + l];
      kf1[0][l] = w * s0;
      kf1[1][l] = w * s1;
    }
#pragma unroll
    for (int j = 0; j < 3; ++j) {
#pragma unroll
      for (int l = 0; l < 3; ++l) {
        float t0 = 0.f, t1 = 0.f;
#pragma unroll
        for (int k = 0; k < 3; ++k) {
          const float w = wjL[224 + li * 27 + j * 9 + l * 3 + k];
          t0 += w * f1v[k];
          t1 += w * f1v[3 + k];
        }
        kf1[0][l] += R11[j * 4 + 0] * t0 + R11[j * 4 + 1] * t1;
        kf1[1][l] += R11[j * 4 + 2] * t0 + R11[j * 4 + 3] * t1;
      }
    }
  }
  // dot with q[v] computed on the fly: q[o][0]=wq0[o]·f0v, q[o][1+k]=wq1[o]·f1v[:,k]
  float dot = 0.f;
#pragma unroll
  for (int o = 0; o < 2; ++o) {
    const float q0 = wq[o * 2 + 0] * f0v0 + wq[o * 2 + 1] * f0v1;
    dot += q0 * (o == 0 ? kf00 : kf01);
#pragma unroll
    for (int k = 0; k < 3; ++k) {
      const float q1 = wq[4 + o * 2 + 0] * f1v[k] + wq[4 + o * 2 + 1] * f1v[3 + k];
      dot += q1 * kf1[o][k];
    }
  }

  if (eo < E) {
    dotbuf[eo] = dot;
    atomicMaxFloat(&mbuf[vo], dot);
  }
}

__global__ void sumexp_kernel(const int* __restrict__ vi,
                              const float* __restrict__ dotbuf,
                              const float* __restrict__ mbuf,
                              float* __restrict__ sbuf, int E) {
  const int e = blockIdx.x * blockDim.x + threadIdx.x;
  if (e >= E) return;
  const int v = vi[e];
  atomicAdd(&sbuf[v], expf(dotbuf[e] - mbuf[v]));
}

__global__ void finalize_kernel(const int* __restrict__ vi,
                                const float* __restrict__ dotbuf,
                                const float* __restrict__ mbuf,
                                const float* __restrict__ sbuf,
                                float* __restrict__ out, int E) {
  const int e = blockIdx.x * blockDim.x + threadIdx.x;
  if (e >= E) return;
  const int v = vi[e];
  const float lse = mbuf[v] + logf(sbuf[v]);
  out[e] = expf(dotbuf[e] - lse);
}

extern "C" void kernel_launch(void* const* d_in, const int* in_sizes, int n_in,
                              void* d_out, int out_size, void* d_ws,
                              size_t ws_size, hipStream_t stream) {
  const float* f0 = (const float*)d_in[0];
  const float* f1 = (const float*)d_in[1];
  const float* dist = (const float*)d_in[2];
  const int* u = (const int*)d_in[3];
  const int* v = (const int*)d_in[4];
  const float* wq = (const float*)d_in[5];
  const float* wj00 = (const float*)d_in[6];
  const float* wj01 = (const float*)d_in[7];
  const float* wj10 = (const float*)d_in[8];
  const float* wj11 = (const float*)d_in[9];
  const float* rW1 = (const float*)d_in[10];
  const float* rb1 = (const float*)d_in[11];
  const float* g1 = (const float*)d_in[12];
  const float* be1 = (const float*)d_in[13];
  const float* rW2 = (const float*)d_in[14];
  const float* rb2 = (const float*)d_in[15];
  const float* g2 = (const float*)d_in[16];
  const float* be2 = (const float*)d_in[17];
  const float* W300 = (const float*)d_in[18];
  const float* b300 = (const float*)d_in[19];
  const float* W301 = (const float*)d_in[20];
  const float* b301 = (const float*)d_in[21];
  const float* W310 = (const float*)d_in[22];
  const float* b310 = (const float*)d_in[23];
  const float* W311 = (const float*)d_in[24];
  const float* b311 = (const float*)d_in[25];

  const int N = in_sizes[0] / 2;  // f0 is [N,2,1]
  const int E = in_sizes[2];      // dist is [E]

  float* mbuf = (float*)d_ws;      // [N] segment max
  float* sbuf = mbuf + N;          // [N] segment sum(exp)
  float* dotbuf = sbuf + N;        // [E] per-edge logits
  float* out = (float*)d_out;

  init_kernel<<<(N + 255) / 256, 256, 0, stream>>>(mbuf, sbuf, N);
  edge_dot_kernel<<<(E + 255) / 256, 256, 0, stream>>>(
      f0, f1, dist, u, v, wq, wj00, wj01, wj10, wj11, rW1, rb1, g1, be1, rW2,
      rb2, g2, be2, W300, b300, W301, b301, W310, b310, W311, b311, dotbuf,
      mbuf, E);
  sumexp_kernel<<<(E + 255) / 256, 256, 0, stream>>>(v, dotbuf, mbuf, sbuf, E);
  finalize_kernel<<<(E + 255) / 256, 256, 0, stream>>>(v, dotbuf, mbuf, sbuf,
                                                       out, E);
}